// RNN_17815524344594
// MI455X (gfx1250) — compile-verified
//
#include <hip/hip_runtime.h>
#include <hip/hip_bf16.h>
#include <math.h>
#include <cstdint>

// Problem: B=64, T=512, D=512, H=512 (fp32 end-to-end)
#define BB 64
#define TT 512
#define DD 512
#define HH 512

typedef __attribute__((ext_vector_type(2))) float    v2f;
typedef __attribute__((ext_vector_type(8))) float    v8f;
typedef __attribute__((ext_vector_type(4))) unsigned v4u;
typedef __attribute__((ext_vector_type(8))) int      v8i_;
typedef __attribute__((ext_vector_type(4))) int      v4i_;

#if defined(__has_builtin)
#if __has_builtin(__builtin_amdgcn_tensor_load_to_lds)
#define TDM_OK 1
#endif
#endif
#ifndef TDM_OK
#define TDM_OK 0
#endif

#if TDM_OK
// 2-D fp32 tile DMA (global -> LDS) through the Tensor Data Mover.
// D# packing per CDNA5 ISA (2-D: groups 2/3 zero).
// pad_int_code 4 => pad every 32 DWORDs; pad_amt_code 3 => 4 DWORDs pad, i.e.
// rows land at a 36-float pitch in LDS (the conflict-free fragment layout).
// This toolchain exposes the 6-arg builtin: (u32x4, i32x8, i32x4, i32x4, i32x8, cpol).
__device__ __forceinline__ void tdm_load_2d(const void* gptr, void* lptr,
                                            unsigned tile_w, unsigned tile_h,
                                            unsigned row_stride_elems,
                                            unsigned tensor_w, unsigned tensor_h,
                                            unsigned pad_int_code,
                                            unsigned pad_amt_code) {
  unsigned long long ga = (unsigned long long)(uintptr_t)gptr;
  unsigned la = (unsigned)(uintptr_t)lptr;       // low 32 bits = LDS byte offset
  v4u g0 = {};
  g0[0] = 1u;                                    // count=1, user descriptor
  g0[1] = la;                                    // lds_addr
  g0[2] = (unsigned)(ga & 0xFFFFFFFFu);          // global_addr[31:0]
  g0[3] = (unsigned)((ga >> 32) & 0x01FFFFFFu)   // global_addr[56:32]
          | (2u << 30);                          // type=2 ("image")
  v8i_ g1 = {};
  g1[0] = (int)((2u << 16)                       // data_size = 4 bytes
                | (1u << 20)                     // pad_enable
                | (pad_int_code << 22) | (pad_amt_code << 25));
  g1[1] = (int)((tensor_w & 0xFFFFu) << 16);                               // tensor_dim0[15:0]
  g1[2] = (int)(((tensor_w >> 16) & 0xFFFFu) | ((tensor_h & 0xFFFFu) << 16)); // dim0 hi | dim1 lo
  g1[3] = (int)(((tensor_h >> 16) & 0xFFFFu) | ((tile_w & 0xFFFFu) << 16));   // dim1 hi | tile_dim0
  g1[4] = (int)(tile_h & 0xFFFFu);               // tile_dim1 (tile_dim2 = 0)
  g1[5] = (int)row_stride_elems;                 // tensor_dim0_stride[31:0]
  g1[6] = 0;
  g1[7] = 0;
  v4i_ gz4 = {};
  v8i_ gz8 = {};
  __builtin_amdgcn_tensor_load_to_lds(g0, g1, gz4, gz4, gz8, 0);
}
#endif

// ---------------------------------------------------------------------------
// Phase 1: g = tanh(X @ W_ih^T), written into the f_seq region of d_out.
// Tile: 128(M) x 64(N) per 256-thread block (8 waves, one 16x64 strip each).
// K staged in 32-wide chunks into double-buffered LDS by the TDM, pipelined
// one chunk ahead (TENSORcnt completes in order per wave, so waiting <=2
// means "previous chunk's A+B pair landed"). V_WMMA_F32_16X16X4_F32 core.
// ---------------------------------------------------------------------------
#define MT 128
#define NT 64
#define KC 32
#define LDA 36   // 36*4=144 B rows: 16B aligned, conflict-free frag reads
#define LDB 36

__global__ __launch_bounds__(256) void in_gemm_tanh(const float* __restrict__ X,
                                                    const float* __restrict__ Wih,
                                                    float* __restrict__ G) {
  __shared__ float As[2][MT * LDA];   // 2 x 18432 B
  __shared__ float Bs[2][NT * LDB];   // 2 x  9216 B
  const int tid  = threadIdx.x;
  const int lane = tid & 31;
  const int wv   = tid >> 5;                 // 0..7 -> 16-row strip
  const int m0   = blockIdx.y * MT;
  const int n0   = blockIdx.x * NT;

  const int mfrag = wv * 16 + (lane & 15);   // A fragment row (local)
  const int khalf = (lane >> 4) * 2;         // half-wave K split
  const int nlane = lane & 15;

  v8f acc[4] = {v8f{}, v8f{}, v8f{}, v8f{}};

#if TDM_OK
  // Prologue: DMA chunk 0. Each wave moves its own disjoint slice (TDM
  // ignores EXEC; all waves execute; slices don't overlap). The TDM's LDS
  // padding produces the 36-float row pitch directly.
  tdm_load_2d(X + (size_t)(m0 + wv * 16) * DD, As[0] + (wv * 16) * LDA,
              KC, 16, DD, DD, BB * TT, 4, 3);
  tdm_load_2d(Wih + (size_t)(n0 + wv * 8) * DD, Bs[0] + (wv * 8) * LDB,
              KC, 8, DD, DD, HH, 4, 3);
#endif

  for (int kk = 0; kk < DD; kk += KC) {
    const int p = (kk / KC) & 1;
#if TDM_OK
    if (kk + KC < DD) {
      // Issue next chunk into the other buffer, then wait only for the
      // *older* pair (<=2 outstanding) so DMA overlaps this chunk's WMMAs.
      tdm_load_2d(X + (size_t)(m0 + wv * 16) * DD + kk + KC,
                  As[p ^ 1] + (wv * 16) * LDA, KC, 16, DD, DD, BB * TT, 4, 3);
      tdm_load_2d(Wih + (size_t)(n0 + wv * 8) * DD + kk + KC,
                  Bs[p ^ 1] + (wv * 8) * LDB, KC, 8, DD, DD, HH, 4, 3);
      __builtin_amdgcn_s_wait_tensorcnt(2);
    } else {
      __builtin_amdgcn_s_wait_tensorcnt(0);
    }
#else
    // Fallback: per-thread float4 staging (single-buffer semantics).
#pragma unroll
    for (int i = 0; i < 4; ++i) {
      int idx = tid + i * 256;               // 0..1023 float4 slots
      int r = idx >> 3, c4 = (idx & 7) * 4;
      *(float4*)(As[p] + r * LDA + c4) =
          *(const float4*)(X + (size_t)(m0 + r) * DD + kk + c4);
    }
#pragma unroll
    for (int i = 0; i < 2; ++i) {
      int idx = tid + i * 256;               // 0..511 float4 slots
      int r = idx >> 3, c4 = (idx & 7) * 4;
      *(float4*)(Bs[p] + r * LDB + c4) =
          *(const float4*)(Wih + (size_t)(n0 + r) * DD + kk + c4);
    }
    if (kk + KC < DD)
      __builtin_prefetch(X + (size_t)(m0 + (tid >> 1)) * DD + kk + KC, 0, 1);
#endif
    __syncthreads();

#pragma unroll
    for (int k4 = 0; k4 < KC; k4 += 4) {
      v2f a = *(const v2f*)(As[p] + mfrag * LDA + k4 + khalf);
#pragma unroll
      for (int nt = 0; nt < 4; ++nt) {
        v2f b = *(const v2f*)(Bs[p] + (nt * 16 + nlane) * LDB + k4 + khalf);
        acc[nt] = __builtin_amdgcn_wmma_f32_16x16x4_f32(
            false, a, false, b, (short)0, acc[nt], false, false);
      }
    }
    __syncthreads();  // buffer p free for the TDM issue 2 chunks ahead
  }

  // Epilogue: tanh + store. VGPR r -> rows (mbase+r); N = lane&15.
  const int mbase = m0 + wv * 16 + ((lane >> 4) ? 8 : 0);
#pragma unroll
  for (int nt = 0; nt < 4; ++nt) {
    const int n = n0 + nt * 16 + nlane;
#pragma unroll
    for (int r = 0; r < 8; ++r) {
      G[(size_t)(mbase + r) * HH + n] = tanhf(acc[nt][r]);
    }
  }
}

// ---------------------------------------------------------------------------
// Phase 2: persistent scan kernel. 32 blocks x 128 threads (4 waves).
// Block b owns output columns [16b,16b+16); its W_hh panel (16x512) lives in
// LDS for all 512 steps. Wave w owns batch rows [16w,16w+16). h ping-pongs
// through L2 (128 KB). Software grid barrier between steps.
// ---------------------------------------------------------------------------
#define SCAN_BLOCKS 32
#define LDW 524    // 524*4=2096 B rows: 16B aligned, conflict-free frag reads

__device__ __forceinline__ void grid_barrier(unsigned* cnt, unsigned* gen,
                                             unsigned nb) {
  __syncthreads();
  if (threadIdx.x == 0) {
    __threadfence();
    volatile unsigned* vgen = (volatile unsigned*)gen;
    unsigned g = *vgen;
    if (atomicAdd(cnt, 1u) == nb - 1u) {
      *(volatile unsigned*)cnt = 0u;
      __threadfence();
      *vgen = g + 1u;
    } else {
      while (*vgen == g) { __builtin_amdgcn_s_sleep(2); }
      __threadfence();
    }
  }
  __syncthreads();
}

__global__ __launch_bounds__(128) void rnn_scan(const float* __restrict__ Whh,
                                                const int* __restrict__ lengths,
                                                float* __restrict__ F,      // d_out
                                                float* __restrict__ hbuf,   // 2 x 64*512
                                                unsigned* __restrict__ bar) {
  __shared__ float Wp[16 * LDW];  // 33536 B
  const int tid  = threadIdx.x;
  const int lane = tid & 31;
  const int wv   = tid >> 5;
  const int n0   = blockIdx.x * 16;

  // Stage W_hh panel: B[k][n] = W_hh[n][k]; rows n0..n0+15, all K.
  for (int i = tid; i < 16 * (DD / 4); i += 128) {
    int r = i >> 7, c4 = (i & 127) * 4;
    *(float4*)(Wp + r * LDW + c4) =
        *(const float4*)(Whh + (size_t)(n0 + r) * HH + c4);
  }
  __syncthreads();

  const int m0    = wv * 16;
  const int mfrag = m0 + (lane & 15);
  const int khalf = (lane >> 4) * 2;
  const int nlane = lane & 15;
  const int nglob = n0 + nlane;
  const int mbase = m0 + ((lane >> 4) ? 8 : 0);

  int lenr[8];
#pragma unroll
  for (int r = 0; r < 8; ++r) lenr[r] = lengths[mbase + r];

  for (int t = 0; t < TT; ++t) {
    const float* hp = hbuf + (size_t)(t & 1) * (BB * HH);
    float*       hn = hbuf + (size_t)((t + 1) & 1) * (BB * HH);

    v8f acc = v8f{};
#pragma unroll 8
    for (int k = 0; k < HH; k += 4) {
      v2f a = *(const v2f*)(hp + (size_t)mfrag * HH + k + khalf);   // A = h_prev
      v2f b = *(const v2f*)(Wp + nlane * LDW + k + khalf);          // B = W_hh^T
      acc = __builtin_amdgcn_wmma_f32_16x16x4_f32(
          false, a, false, b, (short)0, acc, false, false);
    }

#pragma unroll
    for (int r = 0; r < 8; ++r) {
      const int m = mbase + r;
      const size_t gi = ((size_t)m * TT + t) * HH + nglob;  // g in-place in F
      float g    = F[gi];
      float hold = hp[(size_t)m * HH + nglob];
      float hnew = g + (1.0f - g * g) * acc[r];
      float h    = (t < lenr[r]) ? hnew : hold;
      F[gi] = h;                                  // f_seq[b][t][n]
      hn[(size_t)m * HH + nglob] = h;             // next-step state
      if (t == TT - 1)
        F[(size_t)BB * TT * HH + (size_t)m * HH + nglob] = h;  // last hidden
    }
    grid_barrier(bar, bar + 1, SCAN_BLOCKS);
  }
}

// ---------------------------------------------------------------------------
extern "C" void kernel_launch(void* const* d_in, const int* in_sizes, int n_in,
                              void* d_out, int out_size, void* d_ws, size_t ws_size,
                              hipStream_t stream) {
  const float* x       = (const float*)d_in[0];  // [B,T,D]
  const int*   lengths = (const int*)d_in[1];    // [B]
  const float* Wih     = (const float*)d_in[2];  // [H,D]
  const float* Whh     = (const float*)d_in[3];  // [H,H]
  float* F = (float*)d_out;                      // [B*T*H] f_seq + [B*H] last

  unsigned* bar  = (unsigned*)d_ws;                      // 2 counters
  float*    hbuf = (float*)((char*)d_ws + 256);          // 2 x 64*512 floats
  const size_t init_bytes = 256 + (size_t)2 * BB * HH * sizeof(float);
  (void)hipMemsetAsync(d_ws, 0, init_bytes, stream);     // h0 = 0, barrier = 0

  // Phase 1: g = tanh(X W_ih^T) into the f_seq region of d_out.
  in_gemm_tanh<<<dim3(HH / NT, (BB * TT) / MT), 256, 0, stream>>>(x, Wih, F);

  // Phase 2: persistent masked scan, overwrites g with h per step.
  rnn_scan<<<SCAN_BLOCKS, 128, 0, stream>>>(Whh, lengths, F, hbuf, bar);
}